// GAT_36086315221437
// MI455X (gfx1250) — compile-verified
//
#include <hip/hip_runtime.h>

#define DD 128
#define NEG_SLOPE 0.2f

typedef float v2f __attribute__((ext_vector_type(2)));
typedef float v8f __attribute__((ext_vector_type(8)));

// ---------------------------------------------------------------- helpers
__device__ __forceinline__ unsigned enc_f32(float v) {
  unsigned u = __float_as_uint(v);
  return (u & 0x80000000u) ? ~u : (u | 0x80000000u);
}

// ---------------------------------------------------------------- init
__global__ __launch_bounds__(256) void k_init(float* __restrict__ out,
                                              float* __restrict__ denom,
                                              unsigned* __restrict__ m_enc,
                                              int n_nodes, long out4) {
  long i = (long)blockIdx.x * 256 + threadIdx.x;
  if (i < out4) ((float4*)out)[i] = make_float4(0.f, 0.f, 0.f, 0.f);
  if (i < n_nodes) { denom[i] = 0.f; m_enc[i] = 0u; }
}

// ---------------------------------------------------------------- h = x @ W^T (WMMA f32 16x16x4)
// One wave computes a 16-row x 128-col tile of h. W (128x128 f32) staged in LDS.
__global__ __launch_bounds__(256) void k_gemm(const float* __restrict__ x,
                                              const float* __restrict__ W,
                                              float* __restrict__ h,
                                              int n_nodes) {
  __shared__ float Ws[DD * DD]; // 64 KB
  const int tid = threadIdx.x;
  for (int i = tid * 4; i < DD * DD; i += 256 * 4)
    *(float4*)&Ws[i] = *(const float4*)&W[i];
  __syncthreads();

  const int lane = tid & 31;
  const int l16  = lane & 15;
  const int lhi  = lane >> 4;                        // 0 or 1
  const long rowBase = (long)blockIdx.x * 128 + (tid >> 5) * 16;

  // A fragment source row (clamped for tail block; stores are guarded)
  long arow = rowBase + l16;
  if (arow > (long)n_nodes - 1) arow = (long)n_nodes - 1;
  const float* xrow = x + arow * DD + 2 * lhi;

  v8f acc[8] = {};                                   // 8 N-tiles of 16 cols

  #pragma unroll 4
  for (int k = 0; k < DD; k += 4) {
    // A 16x4 f32: lane holds M=l16, K = k + v + 2*lhi  -> contiguous pair
    v2f a = *(const v2f*)(xrow + k);
    #pragma unroll
    for (int t = 0; t < 8; ++t) {
      // B 4x16: lane holds N = 16t+l16, K = k + v + 2*lhi ; B[k][n] = W[n][k]
      v2f b = *(const v2f*)&Ws[(t * 16 + l16) * DD + k + 2 * lhi];
      acc[t] = __builtin_amdgcn_wmma_f32_16x16x4_f32(
          /*neg_a=*/false, a, /*neg_b=*/false, b,
          /*c_mod=*/(short)0, acc[t], /*reuse_a=*/false, /*reuse_b=*/false);
    }
  }

  // C/D layout: vgpr r -> row rowBase + r + 8*lhi, col = t*16 + l16
  #pragma unroll
  for (int t = 0; t < 8; ++t) {
    #pragma unroll
    for (int r = 0; r < 8; ++r) {
      long row = rowBase + r + 8 * lhi;
      if (row < n_nodes) h[row * DD + t * 16 + l16] = acc[t][r];
    }
  }
}

// ---------------------------------------------------------------- e_l = h.aL , e_r = h.aR (wave per node)
__global__ __launch_bounds__(256) void k_el_er(const float* __restrict__ h,
                                               const float* __restrict__ aL,
                                               const float* __restrict__ aR,
                                               float* __restrict__ eL,
                                               float* __restrict__ eR,
                                               int n_nodes) {
  int node = blockIdx.x * 8 + (threadIdx.x >> 5);
  int lane = threadIdx.x & 31;
  if (node >= n_nodes) return;
  float4 hv = *(const float4*)&h[(long)node * DD + lane * 4];
  float4 lv = *(const float4*)&aL[lane * 4];
  float4 rv = *(const float4*)&aR[lane * 4];
  float sl = hv.x * lv.x + hv.y * lv.y + hv.z * lv.z + hv.w * lv.w;
  float sr = hv.x * rv.x + hv.y * rv.y + hv.z * rv.z + hv.w * rv.w;
  #pragma unroll
  for (int m = 16; m >= 1; m >>= 1) {
    sl += __shfl_xor(sl, m, 32);
    sr += __shfl_xor(sr, m, 32);
  }
  if (lane == 0) { eL[node] = sl; eR[node] = sr; }
}

// ---------------------------------------------------------------- edge logits + segment max
__global__ __launch_bounds__(256) void k_edge_e(const float* __restrict__ eL,
                                                const float* __restrict__ eR,
                                                const int* __restrict__ src,
                                                const int* __restrict__ dst,
                                                float* __restrict__ e_edge,
                                                unsigned* __restrict__ m_enc,
                                                int n_edges) {
  int e = blockIdx.x * 256 + threadIdx.x;
  if (e >= n_edges) return;
  int s = src[e], d0 = dst[e];
  float v = eL[s] + eR[d0];
  v = (v >= 0.f) ? v : NEG_SLOPE * v;
  e_edge[e] = v;
  atomicMax(&m_enc[d0], enc_f32(v));
}

// ---------------------------------------------------------------- decode max, clamp non-finite -> 0
__global__ __launch_bounds__(256) void k_m_fix(unsigned* __restrict__ m_enc,
                                               float* __restrict__ m,
                                               int n_nodes) {
  int i = blockIdx.x * 256 + threadIdx.x;
  if (i >= n_nodes) return;
  unsigned ev = m_enc[i];
  unsigned bits = (ev & 0x80000000u) ? (ev & 0x7FFFFFFFu) : ~ev;
  float mm = __uint_as_float(bits);
  if (!(fabsf(mm) < __builtin_inff())) mm = 0.f;   // NaN / +-inf / empty -> 0
  m[i] = mm;
}

// ---------------------------------------------------------------- denom = segment_sum(exp(e - m[dst]))
__global__ __launch_bounds__(256) void k_edge_denom(const float* __restrict__ e_edge,
                                                    const float* __restrict__ m,
                                                    const int* __restrict__ dst,
                                                    float* __restrict__ denom,
                                                    int n_edges) {
  int e = blockIdx.x * 256 + threadIdx.x;
  if (e >= n_edges) return;
  int d0 = dst[e];
  atomicAdd(&denom[d0], expf(e_edge[e] - m[d0]));
}

// ---------------------------------------------------------------- out[dst] += h[src] * alpha  (wave per edge)
__global__ __launch_bounds__(256) void k_edge_msg(const float* __restrict__ h,
                                                  const float* __restrict__ e_edge,
                                                  const float* __restrict__ m,
                                                  const float* __restrict__ denom,
                                                  const int* __restrict__ src,
                                                  const int* __restrict__ dst,
                                                  float* __restrict__ out,
                                                  int n_edges) {
  int edge = blockIdx.x * 8 + (threadIdx.x >> 5);
  int lane = threadIdx.x & 31;
  if (edge >= n_edges) return;
  int s = src[edge], d0 = dst[edge];
  float ex = expf(e_edge[edge] - m[d0]);
  float alpha = ex / fmaxf(denom[d0], 1e-38f);
  float4 hv = *(const float4*)&h[(long)s * DD + lane * 4];
  float* o = out + (long)d0 * DD + lane * 4;
  atomicAdd(o + 0, hv.x * alpha);
  atomicAdd(o + 1, hv.y * alpha);
  atomicAdd(o + 2, hv.z * alpha);
  atomicAdd(o + 3, hv.w * alpha);
}

// ---------------------------------------------------------------- launcher
extern "C" void kernel_launch(void* const* d_in, const int* in_sizes, int n_in,
                              void* d_out, int out_size, void* d_ws, size_t ws_size,
                              hipStream_t stream) {
  const float* x  = (const float*)d_in[0];
  const int*  src = (const int*)d_in[1];
  const int*  dst = (const int*)d_in[2];
  const float* W  = (const float*)d_in[3];
  const float* aL = (const float*)d_in[4];
  const float* aR = (const float*)d_in[5];
  float* out = (float*)d_out;

  const int n_nodes = in_sizes[0] / DD;
  const int n_edges = in_sizes[1];

  // workspace layout
  float* h      = (float*)d_ws;                 // n_nodes * 128
  float* eL     = h + (long)n_nodes * DD;       // n_nodes
  float* eR     = eL + n_nodes;                 // n_nodes
  float* m      = eR + n_nodes;                 // n_nodes (uint view first, then float)
  unsigned* m_enc = (unsigned*)m;
  float* denom  = m + n_nodes;                  // n_nodes
  float* e_edge = denom + n_nodes;              // n_edges

  long out4 = (long)n_nodes * DD / 4;
  long initN = out4 > (long)n_nodes ? out4 : (long)n_nodes;
  dim3 blk(256);

  k_init     <<<(int)((initN + 255) / 256), blk, 0, stream>>>(out, denom, m_enc, n_nodes, out4);
  k_gemm     <<<(n_nodes + 127) / 128,      blk, 0, stream>>>(x, W, h, n_nodes);
  k_el_er    <<<(n_nodes + 7) / 8,          blk, 0, stream>>>(h, aL, aR, eL, eR, n_nodes);
  k_edge_e   <<<(n_edges + 255) / 256,      blk, 0, stream>>>(eL, eR, src, dst, e_edge, m_enc, n_edges);
  k_m_fix    <<<(n_nodes + 255) / 256,      blk, 0, stream>>>(m_enc, m, n_nodes);
  k_edge_denom<<<(n_edges + 255) / 256,     blk, 0, stream>>>(e_edge, m, dst, denom, n_edges);
  k_edge_msg <<<(n_edges + 7) / 8,          blk, 0, stream>>>(h, e_edge, m, denom, src, dst, out, n_edges);
}